// SpatialTransformerNetwork_24343874634010
// MI455X (gfx1250) — compile-verified
//
#include <hip/hip_runtime.h>

// Problem constants (from reference): B=32, H=512, W=512, C=4, scales (0.8, 0.6)
#define B_    32
#define H_    512
#define W_    512
#define C_    4
#define NPIX  (B_ * H_ * W_)      /* 8,388,608 pixels  */
#define NELEM (NPIX * C_)         /* 33,554,432 floats */

// ---------------------------------------------------------------------------
// gfx1250 async Global<->LDS DMA builtins (ASYNCcnt-tracked).
// Probe (round 1 stderr) showed parameter type:
//   '__attribute__((__vector_size__(4 * sizeof(int)))) int __device__ *'
// i.e. pointer-to-int4-vector in the device (global) address space, and the
// LDS side in __shared__ address space.
// ---------------------------------------------------------------------------
#if defined(__gfx1250__) && \
    __has_builtin(__builtin_amdgcn_global_load_async_to_lds_b128) && \
    __has_builtin(__builtin_amdgcn_global_store_async_from_lds_b128) && \
    __has_builtin(__builtin_amdgcn_s_wait_asynccnt)
#define STN_HAVE_ASYNC_LDS 1
#else
#define STN_HAVE_ASYNC_LDS 0
#endif

#if STN_HAVE_ASYNC_LDS
typedef int stn_v4i __attribute__((vector_size(16)));
typedef __attribute__((address_space(1))) stn_v4i* stn_gptr;   // global int4*
typedef __attribute__((address_space(3))) stn_v4i* stn_lptr;   // LDS    int4*
#endif

// ---------------------------------------------------------------------------
// Pass-through copy: out0 = inputs  (128 MB).  Pure bandwidth; route through
// the CDNA5 async LDS DMA path (no VGPR round trip).  Per-wave DMA latency
// is hidden by launching ~32K waves.
// ---------------------------------------------------------------------------
__global__ __launch_bounds__(256)
void stn_copy_async_kernel(const float4* __restrict__ in,
                           float4* __restrict__ out,
                           int n4)
{
#if STN_HAVE_ASYNC_LDS
    __shared__ float4 smem[256];
    const int tid    = threadIdx.x;
    const int stride = gridDim.x * blockDim.x;
    for (int idx = blockIdx.x * blockDim.x + tid; idx < n4; idx += stride) {
        // global -> LDS (async, b128)
        __builtin_amdgcn_global_load_async_to_lds_b128(
            (stn_gptr)(in + idx),
            (stn_lptr)(&smem[tid]),
            0, 0);
        __builtin_amdgcn_s_wait_asynccnt(0);
        // LDS -> global (async, b128)
        __builtin_amdgcn_global_store_async_from_lds_b128(
            (stn_gptr)(out + idx),
            (stn_lptr)(&smem[tid]),
            0, 0);
        __builtin_amdgcn_s_wait_asynccnt(0);   // LDS slot reused next iter
    }
#else
    const int stride = gridDim.x * blockDim.x;
    for (int idx = blockIdx.x * blockDim.x + threadIdx.x; idx < n4; idx += stride) {
        out[idx] = in[idx];
    }
#endif
}

// ---------------------------------------------------------------------------
// Bilinear affine sampler: one thread per output pixel, float4 (b128) taps.
// Exactly replicates the reference math, including the (max-1)=510 scale
// factor and weights computed from *clamped* corner coordinates.
// ---------------------------------------------------------------------------
__global__ __launch_bounds__(256)
void stn_sample_kernel(const float* __restrict__ img,
                       const float* __restrict__ theta,  // (B,2): xt, yt
                       float scale,
                       float* __restrict__ out)
{
    const int p = blockIdx.x * blockDim.x + threadIdx.x;   // pixel id, exact grid
    const int j = p & (W_ - 1);
    const int i = (p >> 9) & (H_ - 1);
    const int b = p >> 18;

    const float xt = theta[2 * b + 0];
    const float yt = theta[2 * b + 1];

    // linspace(-1,1,512): step 2/511
    const float x_t = (float)j * (2.0f / 511.0f) - 1.0f;
    const float y_t = (float)i * (2.0f / 511.0f) - 1.0f;

    const float xs = scale * x_t + xt;
    const float ys = scale * y_t + yt;

    // reference: x = 0.5*((x+1)*(max_x-1)), max_x-1 == 510
    const float px = 0.5f * ((xs + 1.0f) * 510.0f);
    const float py = 0.5f * ((ys + 1.0f) * 510.0f);

    int x0 = (int)floorf(px);
    int y0 = (int)floorf(py);
    int x1 = x0 + 1;
    int y1 = y0 + 1;
    x0 = min(max(x0, 0), W_ - 1);
    x1 = min(max(x1, 0), W_ - 1);
    y0 = min(max(y0, 0), H_ - 1);
    y1 = min(max(y1, 0), H_ - 1);

    const float x0f = (float)x0, x1f = (float)x1;
    const float y0f = (float)y0, y1f = (float)y1;

    const float wa = (x1f - px) * (y1f - py);
    const float wb = (x1f - px) * (py - y0f);
    const float wc = (px - x0f) * (y1f - py);
    const float wd = (px - x0f) * (py - y0f);

    const float4* __restrict__ im = (const float4*)img + (size_t)b * (H_ * W_);
    const float4 Ia = im[y0 * W_ + x0];
    const float4 Ib = im[y1 * W_ + x0];
    const float4 Ic = im[y0 * W_ + x1];
    const float4 Id = im[y1 * W_ + x1];

    float4 r;
    r.x = fmaf(wa, Ia.x, fmaf(wb, Ib.x, fmaf(wc, Ic.x, wd * Id.x)));
    r.y = fmaf(wa, Ia.y, fmaf(wb, Ib.y, fmaf(wc, Ic.y, wd * Id.y)));
    r.z = fmaf(wa, Ia.z, fmaf(wb, Ib.z, fmaf(wc, Ic.z, wd * Id.z)));
    r.w = fmaf(wa, Ia.w, fmaf(wb, Ib.w, fmaf(wc, Ic.w, wd * Id.w)));

    ((float4*)out)[p] = r;
}

// ---------------------------------------------------------------------------
// bound_err: (B,4) = [max(0,|xt1|+0.8-1), max(0,|yt1|+0.8-1),
//                     max(0,|xt2|+0.6-1), max(0,|yt2|+0.6-1)]
// ---------------------------------------------------------------------------
__global__ void stn_bound_err_kernel(const float* __restrict__ t1,
                                     const float* __restrict__ t2,
                                     float* __restrict__ out)
{
    const int b = threadIdx.x;
    if (b < B_) {
        float4 r;
        r.x = fmaxf(0.0f, fabsf(t1[2 * b + 0]) + 0.8f - 1.0f);
        r.y = fmaxf(0.0f, fabsf(t1[2 * b + 1]) + 0.8f - 1.0f);
        r.z = fmaxf(0.0f, fabsf(t2[2 * b + 0]) + 0.6f - 1.0f);
        r.w = fmaxf(0.0f, fabsf(t2[2 * b + 1]) + 0.6f - 1.0f);
        ((float4*)out)[b] = r;
    }
}

// ---------------------------------------------------------------------------
// Launch: d_out layout = [copy (N) | sample0.8 (N) | sample0.6 (N) | berr (128)]
// ---------------------------------------------------------------------------
extern "C" void kernel_launch(void* const* d_in, const int* in_sizes, int n_in,
                              void* d_out, int out_size, void* d_ws, size_t ws_size,
                              hipStream_t stream)
{
    (void)in_sizes; (void)n_in; (void)out_size; (void)d_ws; (void)ws_size;

    const float* inputs = (const float*)d_in[0];
    const float* theta1 = (const float*)d_in[1];
    const float* theta2 = (const float*)d_in[2];
    float* out = (float*)d_out;

    const size_t N = (size_t)NELEM;

    // out0 = inputs (b128 copy, async-LDS DMA path on gfx1250)
    stn_copy_async_kernel<<<4096, 256, 0, stream>>>(
        (const float4*)inputs, (float4*)out, NELEM / 4);

    // out1 = bilinear sample, scale 0.8
    stn_sample_kernel<<<NPIX / 256, 256, 0, stream>>>(
        inputs, theta1, 0.8f, out + N);

    // out2 = bilinear sample, scale 0.6
    stn_sample_kernel<<<NPIX / 256, 256, 0, stream>>>(
        inputs, theta2, 0.6f, out + 2 * N);

    // bound_err (B,4)
    stn_bound_err_kernel<<<1, 32, 0, stream>>>(theta1, theta2, out + 3 * N);
}